// OuterMean_1726576856769
// MI455X (gfx1250) — compile-verified
//
#include <hip/hip_runtime.h>

// Shapes (fixed by the reference): b=2, m=64, i=j=256, d=h=256
#define B_  2
#define M_  64
#define N_  256
#define D_  256
#define H_  256

typedef float v2f __attribute__((ext_vector_type(2)));
typedef float v8f __attribute__((ext_vector_type(8)));

// D(16x16,f32) = A(16x4,f32) x B(4x16,f32) + C   — full-precision CDNA5 WMMA
__device__ __forceinline__ v8f wmma4(v2f a, v2f b, v8f c) {
  return __builtin_amdgcn_wmma_f32_16x16x4_f32(false, a, false, b, (short)0, c,
                                               false, false);
}

// ---------------------------------------------------------------------------
// Kernel 1a: LayerNorm + mask fold.  One wave32 per row (d=256 -> 8 f32/lane).
// xnm[r][k] = ((x-mu)*rsqrt(var+eps)*w + b) * maskf[r],  maskf emitted as f32.
// ---------------------------------------------------------------------------
__global__ __launch_bounds__(256) void ln_mask_kernel(
    const float* __restrict__ x, const unsigned char* __restrict__ mask,
    const float* __restrict__ lnw, const float* __restrict__ lnb,
    float* __restrict__ xnm, float* __restrict__ maskf) {
  int lane = threadIdx.x & 31;
  int wave = threadIdx.x >> 5;
  int row = blockIdx.x * 8 + wave;  // 32768 rows total
  const float* xr = x + (size_t)row * D_;
  float v[8];
  float s = 0.f, s2 = 0.f;
#pragma unroll
  for (int j = 0; j < 8; ++j) {
    v[j] = xr[lane + 32 * j];
    s += v[j];
    s2 += v[j] * v[j];
  }
#pragma unroll
  for (int off = 16; off >= 1; off >>= 1) {
    s += __shfl_xor(s, off, 32);
    s2 += __shfl_xor(s2, off, 32);
  }
  float mu = s * (1.0f / D_);
  float var = s2 * (1.0f / D_) - mu * mu;
  float rs = rsqrtf(var + 1e-5f);
  float mf = mask[row] ? 1.0f : 0.0f;
  float* o = xnm + (size_t)row * D_;
#pragma unroll
  for (int j = 0; j < 8; ++j) {
    int k = lane + 32 * j;
    o[k] = ((v[j] - mu) * rs * lnw[k] + lnb[k]) * mf;
  }
  if (lane == 0) maskf[row] = mf;
}

// ---------------------------------------------------------------------------
// Kernel 1b: scale[b,i,j] = 1 / (m * (sum_m mf[b,m,i]*mf[b,m,j] + eps))
// ---------------------------------------------------------------------------
__global__ __launch_bounds__(256) void scale_kernel(
    const float* __restrict__ maskf, float* __restrict__ scale) {
  int t = blockIdx.x * blockDim.x + threadIdx.x;  // B*N*N = 131072
  int b = t >> 16;
  int ij = t & 65535;
  int i = ij >> 8, j = ij & 255;
  const float* mb = maskf + b * (M_ * N_);
  float c = 0.f;
  for (int m = 0; m < M_; ++m) c += mb[m * N_ + i] * mb[m * N_ + j];
  scale[t] = 1.0f / ((float)M_ * (c + 1e-5f));
}

// ---------------------------------------------------------------------------
// Kernel 2: transposed projections via WMMA:
//   Lt[b][h][m][i] = (xnm @ Wl)[b,m,i,h] + bl[h]*maskf   (same for Rt/Wr/br)
// GEMM view: D[h, r] = sum_k Wl[k][h] * xnm[r][k],  r = flattened (b,m,i).
// 256 threads (8 waves); each wave owns h-tiles {w, w+8} for both L and R.
// xnm B-tile (16 rows x 64 k) staged in padded LDS for transposed reads.
// ---------------------------------------------------------------------------
__global__ __launch_bounds__(256) void proj_kernel(
    const float* __restrict__ xnm, const float* __restrict__ maskf,
    const float* __restrict__ Wl, const float* __restrict__ bl,
    const float* __restrict__ Wr, const float* __restrict__ br,
    float* __restrict__ Lt, float* __restrict__ Rt) {
  __shared__ float xs[16 * 68];  // [r][k] padded: bank-conflict-free B reads
  int lane = threadIdx.x & 31, wave = threadIdx.x >> 5;
  int lh = lane >> 4, ll = lane & 15;
  int r0 = blockIdx.x * 16;  // 2048 blocks
  v8f aL0 = {}, aL1 = {}, aR0 = {}, aR1 = {};
  int h0a = wave * 16, h0b = (wave + 8) * 16;
  for (int kc = 0; kc < 4; ++kc) {  // K in chunks of 64
    __syncthreads();
    {
      int r = threadIdx.x >> 4;
      int kk = (threadIdx.x & 15) * 4;
      const float4 d =
          *(const float4*)(xnm + (size_t)(r0 + r) * D_ + kc * 64 + kk);
      float* p = &xs[r * 68 + kk];
      p[0] = d.x; p[1] = d.y; p[2] = d.z; p[3] = d.w;
    }
    __syncthreads();
    for (int ks = 0; ks < 16; ++ks) {
      int k0 = ks * 4;
      v2f bf;
#pragma unroll
      for (int v = 0; v < 2; ++v) bf[v] = xs[ll * 68 + k0 + v + 2 * lh];
      v2f al0, al1, ar0, ar1;
#pragma unroll
      for (int v = 0; v < 2; ++v) {
        int kr = (kc * 64 + k0 + v + 2 * lh) * H_;
        al0[v] = Wl[kr + h0a + ll];
        al1[v] = Wl[kr + h0b + ll];
        ar0[v] = Wr[kr + h0a + ll];
        ar1[v] = Wr[kr + h0b + ll];
      }
      aL0 = wmma4(al0, bf, aL0);
      aL1 = wmma4(al1, bf, aL1);
      aR0 = wmma4(ar0, bf, aR0);
      aR1 = wmma4(ar1, bf, aR1);
    }
  }
  // Epilogue: bias*mask fold + coalesced stores into Lt/Rt[b][h][m][i]
  int b = r0 / (M_ * N_);
  int mi = r0 % (M_ * N_);
  int m = mi / N_;
  int i0 = mi % N_;
  float mf = maskf[r0 + ll];
#pragma unroll
  for (int v = 0; v < 8; ++v) {
    int hA = h0a + v + 8 * lh;
    int hB = h0b + v + 8 * lh;
    size_t oa = (((size_t)(b * H_ + hA)) * M_ + m) * N_ + i0 + ll;
    size_t ob = (((size_t)(b * H_ + hB)) * M_ + m) * N_ + i0 + ll;
    Lt[oa] = aL0[v] + bl[hA] * mf;
    Lt[ob] = aL1[v] + bl[hB] * mf;
    Rt[oa] = aR0[v] + br[hA] * mf;
    Rt[ob] = aR1[v] + br[hB] * mf;
  }
}

// ---------------------------------------------------------------------------
// Kernel 3: fused einsum('bmih,bmjh->bijh')/scale + (@Wo + bo).
// One workgroup (512 thr / 16 waves) per (b, 16x16 (i,j) tile): 512 WGs.
// Per 32-wide h-chunk:
//   phase1: wave computes outer 16x16 tile for 2 h values (K=m=64 WMMA chain),
//           applies per-(i,j) scale, writes LDS olds[ij][h] (stride-33 pad).
//   phase2: 256x256x32 GEMM A=olds, B=Wo chunk, acc in 128 VGPRs/wave.
// outer never touches HBM; output written once, coalesced, bias fused.
// ---------------------------------------------------------------------------
__global__ __launch_bounds__(512) void fused_outer_kernel(
    const float* __restrict__ Lt, const float* __restrict__ Rt,
    const float* __restrict__ scale, const float* __restrict__ Wo,
    const float* __restrict__ bo, float* __restrict__ out) {
  __shared__ float olds[256 * 33];  // ~34 KB
  int lane = threadIdx.x & 31, wave = threadIdx.x >> 5;  // 16 waves
  int lh = lane >> 4, ll = lane & 15;
  int b = blockIdx.x >> 8;
  int rem = blockIdx.x & 255;
  int i0 = (rem >> 4) * 16;
  int j0 = (rem & 15) * 16;
  // per-lane scale registers: same (i,j) positions for every h
  float sc[8];
#pragma unroll
  for (int v = 0; v < 8; ++v)
    sc[v] = scale[(size_t)b * N_ * N_ + (size_t)(i0 + v + 8 * lh) * N_ + j0 + ll];
  v8f acc[16];
#pragma unroll
  for (int t = 0; t < 16; ++t) acc[t] = (v8f){};
  int d0 = wave * 16;  // phase-2 N-tile per wave
  for (int hc = 0; hc < 8; ++hc) {
    int hbase = hc * 32;
    __syncthreads();
    // ---- phase 1: outer tiles for this h chunk ----
#pragma unroll
    for (int hh = 0; hh < 2; ++hh) {
      int h = hbase + wave * 2 + hh;
      const float* Lb = Lt + (((size_t)(b * H_ + h)) * M_) * N_ + i0;
      const float* Rb = Rt + (((size_t)(b * H_ + h)) * M_) * N_ + j0;
      v8f o = (v8f){};
      for (int ks = 0; ks < 16; ++ks) {  // K = m = 64
        int k0 = ks * 4;
        v2f a, bf;
#pragma unroll
        for (int v = 0; v < 2; ++v) {
          int kr = (k0 + v + 2 * lh) * N_;
          a[v] = Lb[kr + ll];
          bf[v] = Rb[kr + ll];
        }
        o = wmma4(a, bf, o);
      }
      int hloc = h - hbase;
#pragma unroll
      for (int v = 0; v < 8; ++v) {
        int ij = (v + 8 * lh) * 16 + ll;
        olds[ij * 33 + hloc] = o[v] * sc[v];
      }
    }
    __syncthreads();
    // ---- phase 2: accumulate (olds @ Wo) for this chunk ----
    for (int ks = 0; ks < 8; ++ks) {  // K = 32 per chunk
      int k0 = ks * 4;
      v2f bw;
#pragma unroll
      for (int v = 0; v < 2; ++v)
        bw[v] = Wo[(size_t)(hbase + k0 + v + 2 * lh) * D_ + d0 + ll];
#pragma unroll
      for (int mt = 0; mt < 16; ++mt) {
        v2f a;
#pragma unroll
        for (int v = 0; v < 2; ++v)
          a[v] = olds[(mt * 16 + ll) * 33 + k0 + v + 2 * lh];
        acc[mt] = wmma4(a, bw, acc[mt]);
      }
    }
  }
  // Epilogue: bias + coalesced 64B-run stores of out[b,i,j,d]
  float bov = bo[d0 + ll];
#pragma unroll
  for (int mt = 0; mt < 16; ++mt) {
#pragma unroll
    for (int v = 0; v < 8; ++v) {
      int Mrow = mt * 16 + v + 8 * lh;  // flattened (i_local, j_local)
      int il = Mrow >> 4, jl = Mrow & 15;
      size_t oidx =
          (((size_t)b * N_ + (i0 + il)) * N_ + (j0 + jl)) * D_ + d0 + ll;
      out[oidx] = acc[mt][v] + bov;
    }
  }
}

// ---------------------------------------------------------------------------
// Host launcher
// ---------------------------------------------------------------------------
extern "C" void kernel_launch(void* const* d_in, const int* in_sizes, int n_in,
                              void* d_out, int out_size, void* d_ws,
                              size_t ws_size, hipStream_t stream) {
  const float* x = (const float*)d_in[0];
  const unsigned char* mask = (const unsigned char*)d_in[1];  // bool (b,m,i)
  const float* lnw = (const float*)d_in[2];
  const float* lnb = (const float*)d_in[3];
  const float* Wl = (const float*)d_in[4];
  const float* bl = (const float*)d_in[5];
  const float* Wr = (const float*)d_in[6];
  const float* br = (const float*)d_in[7];
  const float* Wo = (const float*)d_in[8];
  const float* bo = (const float*)d_in[9];
  float* ws = (float*)d_ws;
  // workspace layout (floats): xnm 8M, maskf 32K, scale 128K, Lt 8M, Rt 8M
  float* xnm = ws;
  float* maskf = ws + 8388608;
  float* scale = ws + 8421376;
  float* Lt = ws + 8552448;
  float* Rt = ws + 16941056;
  float* out = (float*)d_out;

  ln_mask_kernel<<<4096, 256, 0, stream>>>(x, mask, lnw, lnb, xnm, maskf);
  scale_kernel<<<512, 256, 0, stream>>>(maskf, scale);
  proj_kernel<<<2048, 256, 0, stream>>>(xnm, maskf, Wl, bl, Wr, br, Lt, Rt);
  fused_outer_kernel<<<512, 512, 0, stream>>>(Lt, Rt, scale, Wo, bo, out);
}